// Lstm_Signal_48636209660558
// MI455X (gfx1250) — compile-verified
//
#include <hip/hip_runtime.h>

typedef float v2f __attribute__((ext_vector_type(2)));
typedef float v8f __attribute__((ext_vector_type(8)));

#define TT   2048
#define HID  15
#define TC   128   // x2 staging chunk (columns of t per LDS refill)

__device__ __forceinline__ float fexp2(float x){ return __builtin_amdgcn_exp2f(x); }
__device__ __forceinline__ float frcp (float x){ return __builtin_amdgcn_rcpf(x); }

#if __has_builtin(__builtin_amdgcn_tanhf)
__device__ __forceinline__ float tanh_fast(float x){ return __builtin_amdgcn_tanhf(x); }
#elif __has_builtin(__builtin_amdgcn_tanh_f32)
__device__ __forceinline__ float tanh_fast(float x){ return __builtin_amdgcn_tanh_f32(x); }
#else
// fallback: tanh(x) = 2/(1+exp2(-2x*log2e)) - 1
__device__ __forceinline__ float tanh_fast(float x){ return 2.0f * frcp(1.0f + fexp2(-2.88539008f * x)) - 1.0f; }
#endif
// sigmoid(x) = 0.5*tanh(0.5x) + 0.5
__device__ __forceinline__ float sigm(float x){ return 0.5f * tanh_fast(0.5f * x) + 0.5f; }

__device__ __forceinline__ v8f wmma4(v2f a, v2f b, v8f c){
  // D = A(16x4 f32) * B(4x16 f32) + C(16x16 f32)
  return __builtin_amdgcn_wmma_f32_16x16x4_f32(false, a, false, b, (short)0, c, false, false);
}

// One wave (32 lanes) handles a 16-row batch tile for the whole sequence.
// D/C tile layout: VGPR r, lanes 0-15 -> (M=r, N=lane); lanes 16-31 -> (M=8+r, N=lane-16)
// A layout (16x4):  lane holds row M=lane&15; VGPR0 = K=4j+2*hi, VGPR1 = K=4j+1+2*hi
// B layout (4x16):  lane holds col N=lane&15; VGPR0 = K=4j+2*hi, VGPR1 = K=4j+1+2*hi
__global__ __launch_bounds__(32) void lstm_fused_kernel(
    const float* __restrict__ x2,
    const float* __restrict__ Wih0, const float* __restrict__ Whh0,
    const float* __restrict__ bih0, const float* __restrict__ bhh0,
    const float* __restrict__ Wih1, const float* __restrict__ Whh1,
    const float* __restrict__ bih1, const float* __restrict__ bhh1,
    const float* __restrict__ Wout, const float* __restrict__ bout,
    float* __restrict__ out)
{
  __shared__ float xbuf[16 * TC];  // staged x2 chunk: [row 0..15][t-in-chunk]
  __shared__ float h0b[256];       // h0 tile [m][k]
  __shared__ float h1b[256];       // h1 tile [m][k]

  const int lane = threadIdx.x;
  const int n    = lane & 15;      // tile column (gate idx) / batch row (for A)
  const int hi   = lane >> 4;      // which 16-lane half
  const int b0   = blockIdx.x << 4;

  // ---------------- preload weights into registers (B-operand layout) --------
  // Extended B matrices are [K=16][N=64]; K=15 row carries W_ih0 (x weight) for
  // layer 0 and zeros for layer 1; N-pad columns (n==15) are zero.
  v2f Bw0[4][4], B1h[4][4], B1i[4][4];
  float bs0[4], bs1[4];
  #pragma unroll
  for (int nb = 0; nb < 4; ++nb){
    const int gr = nb * HID + n;                  // gate row, valid when n<15
    bs0[nb] = (n < HID) ? (bih0[gr] + bhh0[gr]) : 0.0f;
    bs1[nb] = (n < HID) ? (bih1[gr] + bhh1[gr]) : 0.0f;
    #pragma unroll
    for (int j = 0; j < 4; ++j){
      const int k0 = 4*j + 2*hi;                  // k0 in {0,2,...,14}; k0+1 may be 15
      float w0x=0.f, w0y=0.f, hx=0.f, hy=0.f, ix=0.f, iy=0.f;
      if (n < HID){
        w0x = Whh0[gr*HID + k0];
        w0y = (k0+1 < HID) ? Whh0[gr*HID + k0 + 1] : Wih0[gr];   // K=15 slot <- x weight
        hx  = Whh1[gr*HID + k0];
        hy  = (k0+1 < HID) ? Whh1[gr*HID + k0 + 1] : 0.0f;
        ix  = Wih1[gr*HID + k0];
        iy  = (k0+1 < HID) ? Wih1[gr*HID + k0 + 1] : 0.0f;
      }
      v2f t0; t0[0]=w0x; t0[1]=w0y; Bw0[nb][j]=t0;
      v2f t1; t1[0]=hx;  t1[1]=hy;  B1h[nb][j]=t1;
      v2f t2; t2[0]=ix;  t2[1]=iy;  B1i[nb][j]=t2;
    }
  }

  // ---------------- state (D-tile layout: value r <-> row m=r+8*hi, col n) ---
  float h0v[8], h1v[8], c0[8], c1[8];
  #pragma unroll
  for (int r = 0; r < 8; ++r){ h0v[r]=0.f; h1v[r]=0.f; c0[r]=0.f; c1[r]=0.f; }

  {  // zero-init the h tiles in LDS (h(-1) = 0)
    const int base = (hi << 7) + n;   // 8*hi*16 + n
    #pragma unroll
    for (int r = 0; r < 8; ++r){ h0b[base + r*16] = 0.0f; h1b[base + r*16] = 0.0f; }
  }

  #pragma unroll 1
  for (int t = 0; t < TT; ++t){
    // ---- refill x staging chunk every TC steps (16 rows x TC floats) ----
    if ((t & (TC-1)) == 0){
      const float* src = x2 + (size_t)(b0 + n) * TT + t + hi * (TC/2);
      float*       dst = &xbuf[n * TC + hi * (TC/2)];
      #pragma unroll
      for (int i = 0; i < TC/8; ++i)
        ((float4*)dst)[i] = ((const float4*)src)[i];
    }

    // ---- layer 0 A operand; K=15 slot (chunk 3, VGPR1, hi lanes) carries x(t)
    v2f A0[4];
    #pragma unroll
    for (int j = 0; j < 4; ++j)
      A0[j] = *(const v2f*)&h0b[n*16 + 4*j + 2*hi];
    {
      float xv = xbuf[n*TC + (t & (TC-1))];   // lane n / n+16 both read x[row n][t]
      if (hi) A0[3][1] = xv;                  // single v_cndmask, no LDS round-trip
    }

    v8f g0[4];
    #pragma unroll
    for (int nb = 0; nb < 4; ++nb){
      v8f acc = {};
      #pragma unroll
      for (int j = 0; j < 4; ++j) acc = wmma4(A0[j], Bw0[nb][j], acc);
      g0[nb] = acc;
    }
    #pragma unroll
    for (int r = 0; r < 8; ++r){
      float ig = sigm     (g0[0][r] + bs0[0]);
      float fg = sigm     (g0[1][r] + bs0[1]);
      float gg = tanh_fast(g0[2][r] + bs0[2]);
      float og = sigm     (g0[3][r] + bs0[3]);
      float c  = fg * c0[r] + ig * gg;
      c0[r]    = c;
      h0v[r]   = og * tanh_fast(c);
    }
    {  // publish new h0 (col-15 junk is harmless: layer-1 K=15 weights are 0)
      const int base = (hi << 7) + n;
      #pragma unroll
      for (int r = 0; r < 8; ++r) h0b[base + r*16] = h0v[r];
    }

    // ---- layer 1: gates = h1(t-1)@Whh1^T + h0(t)@Wih1^T (8-WMMA chain) ----
    v2f Ah1[4], Ah0[4];
    #pragma unroll
    for (int j = 0; j < 4; ++j){
      Ah1[j] = *(const v2f*)&h1b[n*16 + 4*j + 2*hi];
      Ah0[j] = *(const v2f*)&h0b[n*16 + 4*j + 2*hi];
    }
    v8f g1[4];
    #pragma unroll
    for (int nb = 0; nb < 4; ++nb){
      v8f acc = {};
      #pragma unroll
      for (int j = 0; j < 4; ++j) acc = wmma4(Ah1[j], B1h[nb][j], acc);
      #pragma unroll
      for (int j = 0; j < 4; ++j) acc = wmma4(Ah0[j], B1i[nb][j], acc);
      g1[nb] = acc;
    }
    #pragma unroll
    for (int r = 0; r < 8; ++r){
      float ig = sigm     (g1[0][r] + bs1[0]);
      float fg = sigm     (g1[1][r] + bs1[1]);
      float gg = tanh_fast(g1[2][r] + bs1[2]);
      float og = sigm     (g1[3][r] + bs1[3]);
      float c  = fg * c1[r] + ig * gg;
      c1[r]    = c;
      h1v[r]   = og * tanh_fast(c);
    }
    {  // publish new h1 for next step (also used by the head after the loop)
      const int base = (hi << 7) + n;
      #pragma unroll
      for (int r = 0; r < 8; ++r) h1b[base + r*16] = h1v[r];
    }
  }

  // ---------------- output head: out[b, :] = h2_last @ Wout^T + bout ---------
  // h1b holds the final h2 tile. Lane (n, hi) computes batch row n, classes 2hi, 2hi+1.
  {
    const int cls = hi * 2;
    float a0 = bout[cls], a1 = bout[cls + 1];
    #pragma unroll
    for (int k = 0; k < HID; ++k){
      float hv = h1b[n*16 + k];
      a0 += hv * Wout[ cls      * HID + k];
      a1 += hv * Wout[(cls + 1) * HID + k];
    }
    out[(b0 + n) * 4 + cls    ] = a0;
    out[(b0 + n) * 4 + cls + 1] = a1;
  }
}

extern "C" void kernel_launch(void* const* d_in, const int* in_sizes, int n_in,
                              void* d_out, int out_size, void* d_ws, size_t ws_size,
                              hipStream_t stream) {
  const float* x2   = (const float*)d_in[0];
  const float* Wih0 = (const float*)d_in[1];
  const float* Whh0 = (const float*)d_in[2];
  const float* bih0 = (const float*)d_in[3];
  const float* bhh0 = (const float*)d_in[4];
  const float* Wih1 = (const float*)d_in[5];
  const float* Whh1 = (const float*)d_in[6];
  const float* bih1 = (const float*)d_in[7];
  const float* bhh1 = (const float*)d_in[8];
  const float* Wout = (const float*)d_in[9];
  const float* bout = (const float*)d_in[10];
  float* out = (float*)d_out;

  dim3 grid(4096 / 16);   // 256 independent batch tiles
  dim3 block(32);         // one wave per tile
  lstm_fused_kernel<<<grid, block, 0, stream>>>(
      x2, Wih0, Whh0, bih0, bhh0, Wih1, Whh1, bih1, bhh1, Wout, bout, out);
}